// Decoder_82506321756564
// MI455X (gfx1250) — compile-verified
//
#include <hip/hip_runtime.h>
#include <hip/hip_bf16.h>
#include <stdint.h>

typedef float v2f __attribute__((ext_vector_type(2)));
typedef float v4f __attribute__((ext_vector_type(4)));
typedef float v8f __attribute__((ext_vector_type(8)));

#define TT 64
#define BB 64
#define SS 64
#define HH 1024
#define LL 2
#define KCHUNK 32
#define LDSROW 36   // 32 + 4 pad floats; 144B row stride keeps 16B alignment

// ---- CDNA5 async global->LDS path (guarded; falls back to sync staging) ----
#if __has_builtin(__builtin_amdgcn_global_load_async_to_lds_b128)
#define GEMM_ASYNC 1
typedef int v4i __attribute__((vector_size(4 * sizeof(int))));
typedef __attribute__((address_space(1))) v4i* g_v4i_ptr;   // global (AS1)
typedef __attribute__((address_space(3))) v4i* l_v4i_ptr;   // LDS (AS3)
#else
#define GEMM_ASYNC 0
#endif

template <int N>
__device__ __forceinline__ void async_wait()
{
#if __has_builtin(__builtin_amdgcn_s_wait_asynccnt)
    __builtin_amdgcn_s_wait_asynccnt(N);
#else
    asm volatile("s_wait_asynccnt %0" :: "n"(N) : "memory");
#endif
}

__device__ __forceinline__ float sigmoidf_(float x) { return 1.0f / (1.0f + expf(-x)); }

#if GEMM_ASYNC
// Issue one 64x32 chunk (8KB) as 4 async b128 loads per thread (128 threads).
__device__ __forceinline__ void issue_chunk_async(const float* __restrict__ gbase,
                                                  int rowStride, int k0,
                                                  float* lds, int tid)
{
    #pragma unroll
    for (int i = 0; i < 4; ++i) {
        int idx = tid + i * 128;            // 0..511 (64 rows * 8 float4)
        int row = idx >> 3;
        int c4  = (idx & 7) << 2;
        const float* gp = gbase + (size_t)row * rowStride + k0 + c4;
        float* lp = lds + row * LDSROW + c4;
        __builtin_amdgcn_global_load_async_to_lds_b128(
            (g_v4i_ptr)(uintptr_t)gp,
            (l_v4i_ptr)(uint32_t)(uintptr_t)lp,
            0, 0);
    }
}
#endif

// Sync staging fallback (also used when async builtin is unavailable).
__device__ __forceinline__ void load_chunk_sync(const float* __restrict__ gbase,
                                                int rowStride, int k0,
                                                float* lds, int tid)
{
    #pragma unroll
    for (int i = 0; i < 4; ++i) {
        int idx = tid + i * 128;
        int row = idx >> 3;
        int c4  = (idx & 7) << 2;
        v4f v = *(const v4f*)(gbase + (size_t)row * rowStride + k0 + c4);
        *(v4f*)(&lds[row * LDSROW + c4]) = v;
    }
}

// 8 k-steps of 4, WMMA over 4 M-tiles per operand pair.
template <bool HAS2>
__device__ __forceinline__ void wmma_chunk(const float* __restrict__ As0,
                                           const float* __restrict__ Ws0,
                                           const float* __restrict__ As1,
                                           const float* __restrict__ Ws1,
                                           v8f (&acc)[4], int nloc, int kh, int mrow)
{
    #pragma unroll
    for (int kk = 0; kk < KCHUNK; kk += 4) {
        v2f b0;
        b0.x = Ws0[nloc * LDSROW + kk + kh];
        b0.y = Ws0[nloc * LDSROW + kk + kh + 1];
        #pragma unroll
        for (int mt = 0; mt < 4; ++mt) {
            v2f a;
            a.x = As0[(mt * 16 + mrow) * LDSROW + kk + kh];
            a.y = As0[(mt * 16 + mrow) * LDSROW + kk + kh + 1];
            acc[mt] = __builtin_amdgcn_wmma_f32_16x16x4_f32(
                false, a, false, b0, (short)0, acc[mt], false, false);
        }
        if constexpr (HAS2) {
            v2f b1;
            b1.x = Ws1[nloc * LDSROW + kk + kh];
            b1.y = Ws1[nloc * LDSROW + kk + kh + 1];
            #pragma unroll
            for (int mt = 0; mt < 4; ++mt) {
                v2f a;
                a.x = As1[(mt * 16 + mrow) * LDSROW + kk + kh];
                a.y = As1[(mt * 16 + mrow) * LDSROW + kk + kh + 1];
                acc[mt] = __builtin_amdgcn_wmma_f32_16x16x4_f32(
                    false, a, false, b1, (short)0, acc[mt], false, false);
            }
        }
    }
}

// ---------------------------------------------------------------------------
// C[64 x Ntot] = A0[64 x K] * W0^T (+ A1[64 x K] * W1^T) (+ bias0 + bias1), opt tanh.
// W rows (length K) live at W + n*ldW. Block = 128 threads (4 waves), covers 64 cols.
// FP32 WMMA 16x16x4 with async double-buffered LDS staging.
// ---------------------------------------------------------------------------
template <bool HAS2>
__global__ __launch_bounds__(128)
void gemm_dual_wmma(const float* __restrict__ A0, const float* __restrict__ W0,
                    const float* __restrict__ A1, const float* __restrict__ W1,
                    const float* __restrict__ bias0, const float* __restrict__ bias1,
                    float* __restrict__ C, int K, int ldW, int Ntot, int act)
{
#if GEMM_ASYNC
    constexpr int NBUF = 2;
    constexpr int LPER = HAS2 ? 16 : 8;   // async instrs per wave per chunk
#else
    constexpr int NBUF = 1;
#endif
    __shared__ float As0[NBUF * 64 * LDSROW];
    __shared__ float Ws0[NBUF * 64 * LDSROW];
    __shared__ float As1[HAS2 ? NBUF * 64 * LDSROW : 1];
    __shared__ float Ws1[HAS2 ? NBUF * 64 * LDSROW : 1];
    constexpr int BUFSZ = 64 * LDSROW;

    const int tid  = threadIdx.x;
    const int lane = tid & 31;
    const int wave = tid >> 5;
    const int nb   = blockIdx.x * 64;

    v8f acc[4];
    const v8f vz = {};
    acc[0] = vz; acc[1] = vz; acc[2] = vz; acc[3] = vz;

    const int nloc = wave * 16 + (lane & 15);  // column within block strip
    const int kh   = (lane < 16) ? 0 : 2;      // K-halves split across lane halves
    const int mrow = (lane & 15);

    const float* Wb0 = W0 + (size_t)nb * ldW;
    const float* Wb1 = HAS2 ? (W1 + (size_t)nb * ldW) : nullptr;

#if GEMM_ASYNC
    // ---- async double-buffered pipeline ----
    issue_chunk_async(A0,  K,   0, As0, tid);
    issue_chunk_async(Wb0, ldW, 0, Ws0, tid);
    if constexpr (HAS2) {
        issue_chunk_async(A1,  K,   0, As1, tid);
        issue_chunk_async(Wb1, ldW, 0, Ws1, tid);
    }
    int cur = 0;
    for (int k0 = 0; k0 < K; k0 += KCHUNK) {
        const bool more = (k0 + KCHUNK) < K;
        if (more) {
            const int nxt = cur ^ 1;
            issue_chunk_async(A0,  K,   k0 + KCHUNK, As0 + nxt * BUFSZ, tid);
            issue_chunk_async(Wb0, ldW, k0 + KCHUNK, Ws0 + nxt * BUFSZ, tid);
            if constexpr (HAS2) {
                issue_chunk_async(A1,  K,   k0 + KCHUNK, As1 + nxt * BUFSZ, tid);
                issue_chunk_async(Wb1, ldW, k0 + KCHUNK, Ws1 + nxt * BUFSZ, tid);
            }
            async_wait<LPER>();   // previous chunk complete (in-order), next in flight
        } else {
            async_wait<0>();
        }
        __syncthreads();
        wmma_chunk<HAS2>(As0 + cur * BUFSZ, Ws0 + cur * BUFSZ,
                         As1 + (HAS2 ? cur * BUFSZ : 0),
                         Ws1 + (HAS2 ? cur * BUFSZ : 0),
                         acc, nloc, kh, mrow);
        __syncthreads();          // all reads done before this buffer is re-filled
        cur ^= 1;
    }
#else
    // ---- sync single-buffer fallback ----
    for (int k0 = 0; k0 < K; k0 += KCHUNK) {
        load_chunk_sync(A0,  K,   k0, As0, tid);
        load_chunk_sync(Wb0, ldW, k0, Ws0, tid);
        if constexpr (HAS2) {
            load_chunk_sync(A1,  K,   k0, As1, tid);
            load_chunk_sync(Wb1, ldW, k0, Ws1, tid);
        }
        __syncthreads();
        wmma_chunk<HAS2>(As0, Ws0, As1, Ws1, acc, nloc, kh, mrow);
        __syncthreads();
    }
#endif

    // --- epilogue: bias, optional tanh, store per documented C/D layout ---
    const int nglob = nb + wave * 16 + (lane & 15);
    float bv = 0.0f;
    if (bias0) bv += bias0[nglob];
    if (bias1) bv += bias1[nglob];
    const int mbase = (lane < 16) ? 0 : 8;
    #pragma unroll
    for (int mt = 0; mt < 4; ++mt) {
        #pragma unroll
        for (int r = 0; r < 8; ++r) {
            float v = acc[mt][r] + bv;
            if (act) v = tanhf(v);
            C[(size_t)(mt * 16 + mbase + r) * Ntot + nglob] = v;
        }
    }
}

// ---------------------------------------------------------------------------
// Embedding gather: x[t*B+b][:] = emb[input[t*B+b]][:]
// ---------------------------------------------------------------------------
__global__ __launch_bounds__(256)
void embed_kernel(const int* __restrict__ idx, const float* __restrict__ emb,
                  float* __restrict__ x)
{
    int tb = blockIdx.x;
    int row = idx[tb];
    const v4f* src = (const v4f*)(emb + (size_t)row * HH);
    v4f* dst = (v4f*)(x + (size_t)tb * HH);
    dst[threadIdx.x] = src[threadIdx.x];   // 256 * float4 = 1024 floats
}

// ---------------------------------------------------------------------------
// LSTM cell elementwise (torch gate order i,f,g,o). Updates h,c in place.
// ---------------------------------------------------------------------------
__global__ __launch_bounds__(256)
void lstm_cell_kernel(const float* __restrict__ gates,
                      float* __restrict__ h, float* __restrict__ c)
{
    int i = blockIdx.x * blockDim.x + threadIdx.x;  // 0..B*H-1
    int b = i >> 10, n = i & 1023;
    const float* g = gates + (size_t)b * 4 * HH;
    float ig = sigmoidf_(g[n]);
    float fg = sigmoidf_(g[HH + n]);
    float gg = tanhf(g[2 * HH + n]);
    float og = sigmoidf_(g[3 * HH + n]);
    float cn = fg * c[i] + ig * gg;
    c[i] = cn;
    h[i] = og * tanhf(cn);
}

// ---------------------------------------------------------------------------
// Attention: scores = ctx[b] . q[b], softmax over S, wc = a . ctx[b].
// One block per batch element. ctx[b][s][:] = context[s][b][:].
// ---------------------------------------------------------------------------
__global__ __launch_bounds__(256)
void attn_kernel(const float* __restrict__ q, const float* __restrict__ context,
                 float* __restrict__ wc, float* __restrict__ attn_out)
{
    int b = blockIdx.x;
    int tid = threadIdx.x;
    __shared__ float qs[HH];
    __shared__ float part[256];
    __shared__ float sc[SS];
    __shared__ float red[SS];

    for (int h = tid; h < HH; h += 256) qs[h] = q[(size_t)b * HH + h];
    __syncthreads();

    // scores: 4 threads per source position s
    int s = tid >> 2, l4 = tid & 3;
    const float* row = context + (size_t)s * BB * HH + (size_t)b * HH;
    float p = 0.0f;
    for (int h = l4 * 256; h < l4 * 256 + 256; ++h) p += row[h] * qs[h];
    part[tid] = p;
    __syncthreads();
    if (tid < SS)
        sc[tid] = part[tid * 4] + part[tid * 4 + 1] + part[tid * 4 + 2] + part[tid * 4 + 3];
    __syncthreads();

    // softmax over S=64
    if (tid < SS) red[tid] = sc[tid];
    __syncthreads();
    for (int off = 32; off > 0; off >>= 1) {
        if (tid < off) red[tid] = fmaxf(red[tid], red[tid + off]);
        __syncthreads();
    }
    float mx = red[0];
    __syncthreads();
    if (tid < SS) { sc[tid] = expf(sc[tid] - mx); red[tid] = sc[tid]; }
    __syncthreads();
    for (int off = 32; off > 0; off >>= 1) {
        if (tid < off) red[tid] += red[tid + off];
        __syncthreads();
    }
    float inv = 1.0f / red[0];
    __syncthreads();
    if (tid < SS) {
        sc[tid] *= inv;
        attn_out[(size_t)b * SS + tid] = sc[tid];   // overwritten each step; last wins
    }
    __syncthreads();

    // weighted context
    for (int h = tid; h < HH; h += 256) {
        float acc = 0.0f;
        #pragma unroll 8
        for (int s2 = 0; s2 < SS; ++s2)
            acc += sc[s2] * context[(size_t)s2 * BB * HH + (size_t)b * HH + h];
        wc[(size_t)b * HH + h] = acc;
    }
}

// ---------------------------------------------------------------------------
extern "C" void kernel_launch(void* const* d_in, const int* in_sizes, int n_in,
                              void* d_out, int out_size, void* d_ws, size_t ws_size,
                              hipStream_t stream)
{
    const int*   input   = (const int*)d_in[0];
    const float* h0      = (const float*)d_in[1];
    const float* c0      = (const float*)d_in[2];
    const float* context = (const float*)d_in[3];
    const float* emb     = (const float*)d_in[4];
    const float* w_ih    = (const float*)d_in[5];
    const float* w_hh    = (const float*)d_in[6];
    const float* b_ih    = (const float*)d_in[7];
    const float* b_hh    = (const float*)d_in[8];
    const float* w_in    = (const float*)d_in[9];
    const float* w_out   = (const float*)d_in[10];

    const size_t BH  = (size_t)BB * HH;          // 65536
    const size_t TBE = (size_t)TT * BB * HH;     // 4194304
    const size_t LBH = (size_t)LL * BB * HH;     // 131072

    float* ws    = (float*)d_ws;
    float* x_emb = ws;                  // T*B*E
    float* h     = x_emb + TBE;         // L*B*H
    float* c     = h + LBH;             // L*B*H
    float* gates = c + LBH;             // B*4H
    float* q     = gates + (size_t)BB * 4 * HH;  // B*H
    float* wc    = q + BH;              // B*H

    float* outs = (float*)d_out;        // [T,B,H]
    float* hf   = outs + TBE;           // [L,B,H]
    float* cf   = hf + LBH;             // [L,B,H]
    float* attn = cf + LBH;             // [B,S]

    (void)hipMemcpyAsync(h, h0, LBH * sizeof(float), hipMemcpyDeviceToDevice, stream);
    (void)hipMemcpyAsync(c, c0, LBH * sizeof(float), hipMemcpyDeviceToDevice, stream);

    embed_kernel<<<TT * BB, 256, 0, stream>>>(input, emb, x_emb);

    const size_t Wlayer = (size_t)4 * HH * HH;   // per-layer weight stride
    for (int t = 0; t < TT; ++t) {
        // layer 0: gates = x_t*w_ih0^T + h[0]*w_hh0^T + b
        gemm_dual_wmma<true><<<64, 128, 0, stream>>>(
            x_emb + (size_t)t * BH, w_ih, h, w_hh,
            b_ih, b_hh, gates, HH, HH, 4 * HH, 0);
        lstm_cell_kernel<<<256, 256, 0, stream>>>(gates, h, c);

        // layer 1: input = new h[0], recurrent = old h[1]
        gemm_dual_wmma<true><<<64, 128, 0, stream>>>(
            h, w_ih + Wlayer, h + BH, w_hh + Wlayer,
            b_ih + 4 * HH, b_hh + 4 * HH, gates, HH, HH, 4 * HH, 0);
        lstm_cell_kernel<<<256, 256, 0, stream>>>(gates, h + BH, c + BH);

        // attention: q = h_top * w_in^T
        gemm_dual_wmma<false><<<16, 128, 0, stream>>>(
            h + BH, w_in, nullptr, nullptr, nullptr, nullptr,
            q, HH, HH, HH, 0);
        attn_kernel<<<64, 256, 0, stream>>>(q, context, wc, attn);

        // out = tanh([wc, h_top] * w_out^T)
        gemm_dual_wmma<true><<<16, 128, 0, stream>>>(
            wc, w_out, h + BH, w_out + HH, nullptr, nullptr,
            outs + (size_t)t * BH, HH, 2 * HH, HH, 1);
    }

    (void)hipMemcpyAsync(hf, h, LBH * sizeof(float), hipMemcpyDeviceToDevice, stream);
    (void)hipMemcpyAsync(cf, c, LBH * sizeof(float), hipMemcpyDeviceToDevice, stream);
}